// GAT_30374008717353
// MI455X (gfx1250) — compile-verified
//
#include <hip/hip_runtime.h>
#include <hip/hip_bf16.h>
#include <stdint.h>

typedef __attribute__((ext_vector_type(2))) float v2f;
typedef __attribute__((ext_vector_type(8))) float v8f;

#define NEG_SLOPE 0.2f
#define EPSF 1e-16f

// ---------------- utility fills ----------------
__global__ void fill_u32_kernel(uint32_t* __restrict__ p, uint32_t v, int n) {
    int i = blockIdx.x * blockDim.x + threadIdx.x;
    if (i < n) p[i] = v;
}

// ---------------- fp32 WMMA GEMM: D[N x Nc] = A[N x K] * B[K x Nc] ----------------
// One wave32 computes one 16x16 tile of D; K consumed 4 at a time with
// V_WMMA_F32_16X16X4_F32. K/Nc are compile-time so the K-loop fully unrolls
// into clauses of immediate-offset global loads feeding 16 chained WMMAs.
template <int K, int Nc>
__global__ __launch_bounds__(256) void wmma_gemm_f32(
    const float* __restrict__ A, const float* __restrict__ B,
    float* __restrict__ D, int N) {
    int wave = threadIdx.x >> 5;
    int lane = threadIdx.x & 31;
    int rowTile = blockIdx.x * 8 + wave;
    int m0 = rowTile * 16;
    if (m0 >= N) return;
    int n0 = blockIdx.y * 16;

    int lm  = lane & 15;           // M index (A) / N index (B,D)
    int kg  = (lane >> 4) * 2;     // K sub-group: lanes 0-15 -> K 0,1 ; lanes 16-31 -> K 2,3
    int row = m0 + lm; if (row >= N) row = N - 1;   // clamped load (stores are guarded)

    const float* Abase = A + (size_t)row * K + kg;      // 8B aligned (kg even, K mult of 4)
    const float* Bbase = B + (size_t)kg * Nc + n0 + lm;

    v8f acc = {};
#pragma unroll
    for (int k0 = 0; k0 < K; k0 += 4) {
        v2f a = *(const v2f*)(Abase + k0);              // A[row][k0+kg : k0+kg+1]
        v2f b;
        b[0] = Bbase[(size_t)(k0 + 0) * Nc];            // B[k0+kg  ][n0+lm]
        b[1] = Bbase[(size_t)(k0 + 1) * Nc];            // B[k0+kg+1][n0+lm]
        acc = __builtin_amdgcn_wmma_f32_16x16x4_f32(
            /*neg_a=*/false, a, /*neg_b=*/false, b,
            /*c_mod=*/(short)0, acc, /*reuse_a=*/false, /*reuse_b=*/false);
    }
    // D layout: VGPR r -> lanes 0-15: M=r, lanes 16-31: M=r+8; N = lane&15
    int col = n0 + lm;
    int rb  = (lane >> 4) * 8;
#pragma unroll
    for (int r = 0; r < 8; ++r) {
        int rowD = m0 + rb + r;
        if (rowD < N) D[(size_t)rowD * Nc + col] = acc[r];
    }
}

// ---------------- per-node attention logit halves: as = h.a_src, ad = h.a_dst ----------------
__global__ __launch_bounds__(256) void alpha_kernel(
    const float* __restrict__ h, const float* __restrict__ a_src,
    const float* __restrict__ a_dst, float* __restrict__ as,
    float* __restrict__ ad, int N, int F) {
    int wave = threadIdx.x >> 5, lane = threadIdx.x & 31;
    int n = blockIdx.x * (blockDim.x >> 5) + wave;
    if (n >= N) return;
    float ss = 0.f, sd = 0.f;
    for (int f = lane; f < F; f += 32) {
        float v = h[(size_t)n * F + f];
        ss += v * a_src[f];
        sd += v * a_dst[f];
    }
    for (int off = 16; off > 0; off >>= 1) {
        ss += __shfl_xor(ss, off, 32);
        sd += __shfl_xor(sd, off, 32);
    }
    if (lane == 0) { as[n] = ss; ad[n] = sd; }
}

// ---------------- edge helpers ----------------
__device__ __forceinline__ void edge_sd(const long long* __restrict__ ei,
                                        int E, int idx, int& s, int& d) {
    if (idx < E) { s = (int)ei[idx]; d = (int)ei[(size_t)E + idx]; }
    else         { s = d = idx - E; }   // self-loop
}

__device__ __forceinline__ float leaky(float e) {
    return e > 0.f ? e : NEG_SLOPE * e;
}

// float atomic max via signed-max / unsigned-min trick (init bits = 0xFF800000 = -inf)
__device__ __forceinline__ void atomicMaxF(float* addr, float val) {
    int bits = __float_as_int(val);
    if (bits >= 0) atomicMax((int*)addr, bits);
    else           atomicMin((unsigned int*)addr, (unsigned int)bits);
}

__global__ __launch_bounds__(256) void edge_max_kernel(
    const long long* __restrict__ ei, int E, int N,
    const float* __restrict__ as, const float* __restrict__ ad,
    float* __restrict__ m) {
    int idx = blockIdx.x * blockDim.x + threadIdx.x;
    if (idx >= E + N) return;
    int s, d; edge_sd(ei, E, idx, s, d);
    atomicMaxF(&m[d], leaky(as[s] + ad[d]));
}

__global__ __launch_bounds__(256) void edge_sum_kernel(
    const long long* __restrict__ ei, int E, int N,
    const float* __restrict__ as, const float* __restrict__ ad,
    const float* __restrict__ m, float* __restrict__ denom) {
    int idx = blockIdx.x * blockDim.x + threadIdx.x;
    if (idx >= E + N) return;
    int s, d; edge_sd(ei, E, idx, s, d);
    float e = leaky(as[s] + ad[d]);
    atomicAdd(&denom[d], __expf(e - m[d]));
}

// 32 lanes per edge; lane covers features f, f+32 (coalesced gathers + scatter atomics)
__global__ __launch_bounds__(256) void edge_agg_kernel(
    const long long* __restrict__ ei, int E, int N, int F,
    const float* __restrict__ as, const float* __restrict__ ad,
    const float* __restrict__ m, const float* __restrict__ denom,
    const float* __restrict__ h, float* __restrict__ out) {
    int t = blockIdx.x * blockDim.x + threadIdx.x;
    int idx = t >> 5;
    int lane = t & 31;
    if (idx >= E + N) return;
    int s, d; edge_sd(ei, E, idx, s, d);
    float e = leaky(as[s] + ad[d]);
    float alpha = __expf(e - m[d]) / (denom[d] + EPSF);
    for (int f = lane; f < F; f += 32)
        atomicAdd(&out[(size_t)d * F + f], h[(size_t)s * F + f] * alpha);
}

// ---------------- bias + ReLU (in place) ----------------
__global__ __launch_bounds__(256) void bias_relu_kernel(
    float* __restrict__ x, const float* __restrict__ b, int total, int F) {
    int i = blockIdx.x * blockDim.x + threadIdx.x;
    if (i >= total) return;
    float v = x[i] + b[i % F];
    x[i] = v > 0.f ? v : 0.f;
}

// ---------------- bias + log_softmax over C=32 (wave per node, in place on d_out) ----------------
__global__ __launch_bounds__(256) void logsoftmax_kernel(
    float* __restrict__ out, const float* __restrict__ b, int N) {
    int wave = threadIdx.x >> 5, lane = threadIdx.x & 31;
    int n = blockIdx.x * (blockDim.x >> 5) + wave;
    if (n >= N) return;
    float v = out[(size_t)n * 32 + lane] + b[lane];
    float mx = v;
    for (int off = 16; off > 0; off >>= 1) mx = fmaxf(mx, __shfl_xor(mx, off, 32));
    float ex = __expf(v - mx), sum = ex;
    for (int off = 16; off > 0; off >>= 1) sum += __shfl_xor(sum, off, 32);
    out[(size_t)n * 32 + lane] = v - mx - logf(sum);
}

// ---------------- driver ----------------
extern "C" void kernel_launch(void* const* d_in, const int* in_sizes, int n_in,
                              void* d_out, int out_size, void* d_ws, size_t ws_size,
                              hipStream_t stream) {
    const float*     x    = (const float*)d_in[0];
    const long long* ei   = (const long long*)d_in[1];
    const float*     W1   = (const float*)d_in[2];
    const float*     a1s  = (const float*)d_in[3];
    const float*     a1d  = (const float*)d_in[4];
    const float*     b1   = (const float*)d_in[5];
    const float*     W2   = (const float*)d_in[6];
    const float*     a2s  = (const float*)d_in[7];
    const float*     a2d  = (const float*)d_in[8];
    const float*     b2   = (const float*)d_in[9];
    float*           out  = (float*)d_out;

    const int N  = in_sizes[0] / 64;   // F_in = 64
    const int E  = in_sizes[1] / 2;    // edge_index is (2,E)
    const int ET = E + N;              // + self-loops

    // workspace carve-out: N*(64+64+8) floats (~54 MB for N=100k)
    float* ws   = (float*)d_ws;
    float* h1   = ws;                        // N*64 (reused as h2: N*32)
    float* out1 = h1   + (size_t)N * 64;     // N*64 aggregation accumulator / h1b
    float* as1  = out1 + (size_t)N * 64;
    float* ad1  = as1 + N;
    float* m1   = ad1 + N;
    float* dn1  = m1  + N;
    float* as2  = dn1 + N;
    float* ad2  = as2 + N;
    float* m2   = ad2 + N;
    float* dn2  = m2  + N;

    const int T = 256;
    const uint32_t NEG_INF_BITS = 0xFF800000u;
    auto blks = [](long long n, int t) { return (int)((n + t - 1) / t); };

    // init accumulators / maxima (d_ws and d_out are poisoned; we own their state)
    fill_u32_kernel<<<blks(N, T), T, 0, stream>>>((uint32_t*)m1,  NEG_INF_BITS, N);
    fill_u32_kernel<<<blks(N, T), T, 0, stream>>>((uint32_t*)dn1, 0u, N);
    fill_u32_kernel<<<blks(N, T), T, 0, stream>>>((uint32_t*)m2,  NEG_INF_BITS, N);
    fill_u32_kernel<<<blks(N, T), T, 0, stream>>>((uint32_t*)dn2, 0u, N);
    fill_u32_kernel<<<blks((long long)N * 64, T), T, 0, stream>>>((uint32_t*)out1, 0u, N * 64);
    fill_u32_kernel<<<blks((long long)N * 32, T), T, 0, stream>>>((uint32_t*)out,  0u, N * 32);

    int rowTiles = (N + 15) / 16;

    // ---- layer 1 ----
    {
        dim3 g((rowTiles + 7) / 8, 64 / 16);
        wmma_gemm_f32<64, 64><<<g, T, 0, stream>>>(x, W1, h1, N);
    }
    alpha_kernel<<<blks(N, 8), T, 0, stream>>>(h1, a1s, a1d, as1, ad1, N, 64);
    edge_max_kernel<<<blks(ET, T), T, 0, stream>>>(ei, E, N, as1, ad1, m1);
    edge_sum_kernel<<<blks(ET, T), T, 0, stream>>>(ei, E, N, as1, ad1, m1, dn1);
    edge_agg_kernel<<<blks((long long)ET * 32, T), T, 0, stream>>>(
        ei, E, N, 64, as1, ad1, m1, dn1, h1, out1);
    bias_relu_kernel<<<blks((long long)N * 64, T), T, 0, stream>>>(out1, b1, N * 64, 64);

    // ---- layer 2 (h2 written into h1 buffer) ----
    {
        dim3 g((rowTiles + 7) / 8, 32 / 16);
        wmma_gemm_f32<64, 32><<<g, T, 0, stream>>>(out1, W2, h1, N);
    }
    alpha_kernel<<<blks(N, 8), T, 0, stream>>>(h1, a2s, a2d, as2, ad2, N, 32);
    edge_max_kernel<<<blks(ET, T), T, 0, stream>>>(ei, E, N, as2, ad2, m2);
    edge_sum_kernel<<<blks(ET, T), T, 0, stream>>>(ei, E, N, as2, ad2, m2, dn2);
    edge_agg_kernel<<<blks((long long)ET * 32, T), T, 0, stream>>>(
        ei, E, N, 32, as2, ad2, m2, dn2, h1, out);

    // ---- log_softmax (in place on d_out) ----
    logsoftmax_kernel<<<blks(N, 8), T, 0, stream>>>(out, b2, N);
}